// GAT_83038897701148
// MI455X (gfx1250) — compile-verified
//
#include <hip/hip_runtime.h>

// ---------------------------------------------------------------------------
// GAT 3-layer forward for MI455X (gfx1250, wave32).
// GEMMs on the CDNA5 matrix pipe via V_WMMA_F32_16X16X4_F32 (exact fp32),
// register-blocked 16x64 per wave (A fragment reused across 4 WMMAs/k-step).
// Edge softmax/aggregation is L2-resident atomics (192MB L2 >> 51MB arrays).
// ---------------------------------------------------------------------------

#define GAT_N 50000
#define GAT_E 800000
#define GAT_F 64          // out_feats per head (all layers)

typedef float v2f __attribute__((ext_vector_type(2)));
typedef float v8f __attribute__((ext_vector_type(8)));

// ------------- WMMA fp32 GEMM: C[M,Nw] = A[M,K] * B[K,Nw] ------------------
// One wave per 16x64 C strip (4 x 16x16 WMMA tiles). M %16==0, Nw %64==0,
// K %4==0. A fragment loaded once per k-step, reused by all 4 WMMAs.
__global__ __launch_bounds__(256)
void gat_gemm_wmma(const float* __restrict__ A, const float* __restrict__ B,
                   float* __restrict__ C, int M, int K, int Nw) {
  const int wave = blockIdx.x * 8 + (threadIdx.x >> 5);
  const int lane = threadIdx.x & 31;
  const int tilesN = Nw >> 6;                   // 64-wide strips
  const int tilesM = M >> 4;
  if (wave >= tilesM * tilesN) return;          // whole-wave exit: EXEC all-1 inside
  const int tm = wave / tilesN;
  const int tn = wave - tm * tilesN;
  const int kh = lane >> 4;                     // K-half held by this lane (A & B)
  const int lr = lane & 15;

  const float* Ap = A + (size_t)(tm * 16 + lr) * K + kh * 2;  // row lr, k = k0+kh*2+{0,1}
  const float* Bp = B + (size_t)(tn * 64) + lr;               // strip base, col lr
  v8f acc0 = {}, acc1 = {}, acc2 = {}, acc3 = {};
  for (int k0 = 0; k0 < K; k0 += 4) {
    v2f a = *(const v2f*)(Ap + k0);                           // contiguous in K
    const float* Brow0 = Bp + (size_t)(k0 + kh * 2) * Nw;
    const float* Brow1 = Brow0 + Nw;
    v2f b0, b1, b2, b3;
    b0.x = Brow0[0];  b0.y = Brow1[0];
    b1.x = Brow0[16]; b1.y = Brow1[16];
    b2.x = Brow0[32]; b2.y = Brow1[32];
    b3.x = Brow0[48]; b3.y = Brow1[48];
    acc0 = __builtin_amdgcn_wmma_f32_16x16x4_f32(false, a, false, b0, (short)0, acc0, false, false);
    acc1 = __builtin_amdgcn_wmma_f32_16x16x4_f32(false, a, false, b1, (short)0, acc1, false, false);
    acc2 = __builtin_amdgcn_wmma_f32_16x16x4_f32(false, a, false, b2, (short)0, acc2, false, false);
    acc3 = __builtin_amdgcn_wmma_f32_16x16x4_f32(false, a, false, b3, (short)0, acc3, false, false);
  }
  // C/D layout: VGPR r -> M = r (lanes 0-15) / M = 8+r (lanes 16-31); N = lr
  float* Cp = C + (size_t)(tm * 16 + kh * 8) * Nw + tn * 64 + lr;
#pragma unroll
  for (int r = 0; r < 8; ++r) {
    Cp[(size_t)r * Nw +  0] = acc0[r];
    Cp[(size_t)r * Nw + 16] = acc1[r];
    Cp[(size_t)r * Nw + 32] = acc2[r];
    Cp[(size_t)r * Nw + 48] = acc3[r];
  }
}

// ---------------- per-(node,head) attention dot: out = sum_f h*attn --------
__global__ __launch_bounds__(256)
void gat_node_attn(const float* __restrict__ h, const float* __restrict__ attn,
                   float* __restrict__ out, int N, int H) {
  int idx = blockIdx.x * blockDim.x + threadIdx.x;
  if (idx >= N * H) return;
  int n = idx / H, hh = idx - n * H;
  const float4* hp = (const float4*)(h + ((size_t)n * H + hh) * GAT_F);
  const float4* ap = (const float4*)(attn + (size_t)hh * GAT_F);
  float s = 0.f;
#pragma unroll
  for (int f = 0; f < GAT_F / 4; ++f) {
    float4 x = hp[f], w = ap[f];
    s += x.x * w.x + x.y * w.y + x.z * w.z + x.w * w.w;
  }
  out[idx] = s;
}

// ---------------- generic u32 fill -----------------------------------------
__global__ __launch_bounds__(256)
void gat_fill_u32(unsigned* __restrict__ p, unsigned v, long long n) {
  long long i = (long long)blockIdx.x * blockDim.x + threadIdx.x;
  if (i < n) p[i] = v;
}

// order-preserving float <-> uint key (for atomic segment-max)
__device__ __forceinline__ unsigned f32_key(float x) {
  unsigned u = __float_as_uint(x);
  return (u & 0x80000000u) ? ~u : (u | 0x80000000u);
}
__device__ __forceinline__ float key_f32(unsigned k) {
  return (k & 0x80000000u) ? __uint_as_float(k ^ 0x80000000u) : __uint_as_float(~k);
}
#define KEY_NEG_INF 0x007FFFFFu   // f32_key(-inf)

// ---------------- pass A: leaky-relu logits + segment max ------------------
__global__ __launch_bounds__(256)
void gat_edge_logits(const int* __restrict__ src, const int* __restrict__ dst,
                     const float* __restrict__ el, const float* __restrict__ er,
                     float* __restrict__ ebuf, unsigned* __restrict__ mkey,
                     int E, int H) {
  int idx = blockIdx.x * blockDim.x + threadIdx.x;
  if (idx >= E * H) return;
  int e = idx / H, hh = idx - e * H;
  int s = src[e], d = dst[e];
  float x = el[s * H + hh] + er[d * H + hh];
  x = (x > 0.f) ? x : 0.2f * x;                 // LeakyReLU(0.2)
  ebuf[idx] = x;
  atomicMax(&mkey[d * H + hh], f32_key(x));
}

// ---------------- pass B: exp(e - m[dst]) + segment sum --------------------
__global__ __launch_bounds__(256)
void gat_edge_exp(const int* __restrict__ dst, float* __restrict__ ebuf,
                  const unsigned* __restrict__ mkey, float* __restrict__ ssum,
                  int E, int H) {
  int idx = blockIdx.x * blockDim.x + threadIdx.x;
  if (idx >= E * H) return;
  int e = idx / H, hh = idx - e * H;
  int d = dst[e];
  float m = key_f32(mkey[d * H + hh]);
  float ex = __expf(ebuf[idx] - m);
  ebuf[idx] = ex;                               // overwrite in place
  atomicAdd(&ssum[d * H + hh], ex);
}

// ---------------- pass C: out[dst] += h[src] * alpha  (F=64 fixed) ---------
__global__ __launch_bounds__(256)
void gat_edge_aggregate(const int* __restrict__ src, const int* __restrict__ dst,
                        const float* __restrict__ h, const float* __restrict__ ebuf,
                        const float* __restrict__ ssum, float* __restrict__ out,
                        int E, int H) {
  const int HF = H * GAT_F;
  long long idx = (long long)blockIdx.x * blockDim.x + threadIdx.x;
  if (idx >= (long long)E * HF) return;
  int e = (int)(idx / HF);
  int i = (int)(idx - (long long)e * HF);
  int hh = i >> 6;                              // F == 64
  int s = src[e], d = dst[e];
  float alpha = ebuf[e * H + hh] / ssum[d * H + hh];
  atomicAdd(&out[(size_t)d * HF + i], h[(size_t)s * HF + i] * alpha);
}

// ---------------- epilogues -------------------------------------------------
__global__ __launch_bounds__(256)
void gat_bias_relu(float* __restrict__ x, const float* __restrict__ b,
                   long long n, int HF) {
  long long idx = (long long)blockIdx.x * blockDim.x + threadIdx.x;
  if (idx >= n) return;
  float v = x[idx] + b[(int)(idx % HF)];
  x[idx] = v > 0.f ? v : 0.f;
}

__global__ __launch_bounds__(256)
void gat_bias_out(const float* __restrict__ x, const float* __restrict__ b,
                  float* __restrict__ out, long long n, int HF) {
  long long idx = (long long)blockIdx.x * blockDim.x + threadIdx.x;
  if (idx >= n) return;
  out[idx] = x[idx] + b[(int)(idx % HF)];
}

// ---------------------------------------------------------------------------
extern "C" void kernel_launch(void* const* d_in, const int* in_sizes, int n_in,
                              void* d_out, int out_size, void* d_ws, size_t ws_size,
                              hipStream_t stream) {
  (void)in_sizes; (void)n_in; (void)out_size; (void)ws_size;
  const float* feat = (const float*)d_in[0];
  const int*   src  = (const int*)d_in[1];
  const int*   dst  = (const int*)d_in[2];
  const float* W1  = (const float*)d_in[3];
  const float* al1 = (const float*)d_in[4];
  const float* ar1 = (const float*)d_in[5];
  const float* b1  = (const float*)d_in[6];
  const float* W2  = (const float*)d_in[7];
  const float* al2 = (const float*)d_in[8];
  const float* ar2 = (const float*)d_in[9];
  const float* b2  = (const float*)d_in[10];
  const float* W3  = (const float*)d_in[11];
  const float* al3 = (const float*)d_in[12];
  const float* ar3 = (const float*)d_in[13];
  const float* b3  = (const float*)d_in[14];
  float* out = (float*)d_out;

  const int N = GAT_N, E = GAT_E;

  // workspace carve (256B aligned)
  char* ws = (char*)d_ws;
  size_t off = 0;
  auto carve = [&](size_t bytes) -> void* {
    void* p = ws + off;
    off += (bytes + 255) & ~(size_t)255;
    return p;
  };
  float*    bufA = (float*)carve((size_t)N * 256 * 4);   // GEMM result h
  float*    bufB = (float*)carve((size_t)N * 256 * 4);   // aggregated / next input
  float*    el   = (float*)carve((size_t)N * 4 * 4);
  float*    er   = (float*)carve((size_t)N * 4 * 4);
  unsigned* mkey = (unsigned*)carve((size_t)N * 4 * 4);
  float*    ssum = (float*)carve((size_t)N * 4 * 4);
  float*    ebuf = (float*)carve((size_t)E * 4 * 4);

  auto blocks = [](long long n) { return (unsigned)((n + 255) / 256); };

  // ---------------- layer 1: X=feat [N,128] -> bufA [N,4,64] ----------------
  {
    const int K = 128, Nw = 256, H = 4;
    int waves = (N / 16) * (Nw / 64);
    gat_gemm_wmma<<<(waves + 7) / 8, 256, 0, stream>>>(feat, W1, bufA, N, K, Nw);
    long long nh = (long long)N * H;
    gat_node_attn<<<blocks(nh), 256, 0, stream>>>(bufA, al1, el, N, H);
    gat_node_attn<<<blocks(nh), 256, 0, stream>>>(bufA, ar1, er, N, H);
    gat_fill_u32<<<blocks(nh), 256, 0, stream>>>(mkey, KEY_NEG_INF, nh);
    gat_fill_u32<<<blocks(nh), 256, 0, stream>>>((unsigned*)ssum, 0u, nh);
    long long nout = (long long)N * H * GAT_F;
    gat_fill_u32<<<blocks(nout), 256, 0, stream>>>((unsigned*)bufB, 0u, nout);
    long long eh = (long long)E * H;
    gat_edge_logits<<<blocks(eh), 256, 0, stream>>>(src, dst, el, er, ebuf, mkey, E, H);
    gat_edge_exp<<<blocks(eh), 256, 0, stream>>>(dst, ebuf, mkey, ssum, E, H);
    long long ew = (long long)E * H * GAT_F;
    gat_edge_aggregate<<<blocks(ew), 256, 0, stream>>>(src, dst, bufA, ebuf, ssum, bufB, E, H);
    gat_bias_relu<<<blocks(nout), 256, 0, stream>>>(bufB, b1, nout, H * GAT_F);
  }

  // ---------------- layer 2: X=bufB [N,256] -> bufB [N,4,64] ----------------
  {
    const int K = 256, Nw = 256, H = 4;
    int waves = (N / 16) * (Nw / 64);
    gat_gemm_wmma<<<(waves + 7) / 8, 256, 0, stream>>>(bufB, W2, bufA, N, K, Nw);
    long long nh = (long long)N * H;
    gat_node_attn<<<blocks(nh), 256, 0, stream>>>(bufA, al2, el, N, H);
    gat_node_attn<<<blocks(nh), 256, 0, stream>>>(bufA, ar2, er, N, H);
    gat_fill_u32<<<blocks(nh), 256, 0, stream>>>(mkey, KEY_NEG_INF, nh);
    gat_fill_u32<<<blocks(nh), 256, 0, stream>>>((unsigned*)ssum, 0u, nh);
    long long nout = (long long)N * H * GAT_F;
    gat_fill_u32<<<blocks(nout), 256, 0, stream>>>((unsigned*)bufB, 0u, nout);  // input consumed by GEMM
    long long eh = (long long)E * H;
    gat_edge_logits<<<blocks(eh), 256, 0, stream>>>(src, dst, el, er, ebuf, mkey, E, H);
    gat_edge_exp<<<blocks(eh), 256, 0, stream>>>(dst, ebuf, mkey, ssum, E, H);
    long long ew = (long long)E * H * GAT_F;
    gat_edge_aggregate<<<blocks(ew), 256, 0, stream>>>(src, dst, bufA, ebuf, ssum, bufB, E, H);
    gat_bias_relu<<<blocks(nout), 256, 0, stream>>>(bufB, b2, nout, H * GAT_F);
  }

  // ---------------- layer 3: X=bufB [N,256] -> out [N,64], H=1 --------------
  {
    const int K = 256, Nw = 64, H = 1;
    int waves = (N / 16) * (Nw / 64);
    gat_gemm_wmma<<<(waves + 7) / 8, 256, 0, stream>>>(bufB, W3, bufA, N, K, Nw);
    long long nh = (long long)N * H;
    gat_node_attn<<<blocks(nh), 256, 0, stream>>>(bufA, al3, el, N, H);
    gat_node_attn<<<blocks(nh), 256, 0, stream>>>(bufA, ar3, er, N, H);
    gat_fill_u32<<<blocks(nh), 256, 0, stream>>>(mkey, KEY_NEG_INF, nh);
    gat_fill_u32<<<blocks(nh), 256, 0, stream>>>((unsigned*)ssum, 0u, nh);
    long long nout = (long long)N * H * GAT_F;
    gat_fill_u32<<<blocks(nout), 256, 0, stream>>>((unsigned*)bufB, 0u, nout);
    long long eh = (long long)E * H;
    gat_edge_logits<<<blocks(eh), 256, 0, stream>>>(src, dst, el, er, ebuf, mkey, E, H);
    gat_edge_exp<<<blocks(eh), 256, 0, stream>>>(dst, ebuf, mkey, ssum, E, H);
    long long ew = (long long)E * H * GAT_F;
    gat_edge_aggregate<<<blocks(ew), 256, 0, stream>>>(src, dst, bufA, ebuf, ssum, bufB, E, H);
    // mean over singleton head axis == identity; just add bias into d_out
    gat_bias_out<<<blocks(nout), 256, 0, stream>>>(bufB, b3, out, nout, GAT_F);
  }
}